// Multi_Intergrate_38079180046840
// MI455X (gfx1250) — compile-verified
//
#include <hip/hip_runtime.h>

// ---------------------------------------------------------------------------
// Multi-stage bicubic pyramid for MI455X (gfx1250, wave32).
//   stages 1-3: x2 bicubic upsample + 1x1 mix + ReLU (tiny, VALU)
//   stage 4:    x4 bicubic upsample 256->1024 + batch flip, expressed as
//               V_WMMA_F32_16X16X4_F32 separable-filter GEMMs.
// Output is 256 MB of fp32 stores -> store-bandwidth bound (~11us @ 23.3TB/s);
// everything else stays in L2.
// Workspace layout (d_ws): h64 (1MB) | h128 (4MB) | h256 (16MB)  = 21MB.
// ---------------------------------------------------------------------------

typedef float v2f __attribute__((ext_vector_type(2)));
typedef float v8f __attribute__((ext_vector_type(8)));

// Keys cubic kernel, a = -0.5 (PIL/jax 'cubic')
__device__ __forceinline__ float keys_w(float d) {
    d = fabsf(d);
    if (d < 1.0f)  return (1.5f * d - 2.5f) * d * d + 1.0f;
    if (d < 2.0f)  return ((-0.5f * d + 2.5f) * d - 4.0f) * d + 2.0f;
    return 0.0f;
}

// ---------------------------------------------------------------------------
// Stage kernel: out[B,2n,2n] = relu(w[0]*bicubic2x(prev)[B,2n,2n] + w[1]*cam + b)
// One thread per output pixel; 4x4 separable taps. Tiny vs final stage.
// ---------------------------------------------------------------------------
__global__ void stage_up2_mix_relu(const float* __restrict__ prev,
                                   const float* __restrict__ cam,
                                   const float* __restrict__ w,
                                   const float* __restrict__ bvec,
                                   float* __restrict__ out,
                                   int n, int B)
{
    const int N = n * 2;
    const long long total = (long long)B * N * N;
    long long i = (long long)blockIdx.x * blockDim.x + threadIdx.x;
    if (i >= total) return;

    const int ox = (int)(i % N);
    const int oy = (int)((i / N) % N);
    const int bz = (int)(i / ((long long)N * N));
    const float* src = prev + (size_t)bz * n * n;

    // src coord = (o+0.5)/2 - 0.5 = k + s, s = +/-0.25; taps floor(s)-1..+2
    const int ky = oy >> 1, kx = ox >> 1;
    const float sy = (oy & 1) ? 0.25f : -0.25f;
    const float sx = (ox & 1) ? 0.25f : -0.25f;
    const int ty0 = (oy & 1) ? -1 : -2;
    const int tx0 = (ox & 1) ? -1 : -2;

    float wy[4], wx[4];
#pragma unroll
    for (int j = 0; j < 4; ++j) {
        wy[j] = keys_w(sy - (float)(ty0 + j));
        wx[j] = keys_w(sx - (float)(tx0 + j));
    }

    float acc = 0.0f;
#pragma unroll
    for (int jy = 0; jy < 4; ++jy) {
        const int yy = min(max(ky + ty0 + jy, 0), n - 1);
        const float* rowp = src + (size_t)yy * n;
        float rs = 0.0f;
#pragma unroll
        for (int jx = 0; jx < 4; ++jx) {
            const int xx = min(max(kx + tx0 + jx, 0), n - 1);
            rs = fmaf(wx[jx], rowp[xx], rs);
        }
        acc = fmaf(wy[jy], rs, acc);
    }

    const float val = w[0] * acc + w[1] * cam[i] + bvec[0];
    out[i] = fmaxf(val, 0.0f);
}

// ---------------------------------------------------------------------------
// Final x4 upsample h256[B,256,256] -> out[B,1024,1024] (batch-flipped),
// as separable 16x8 filter GEMMs on V_WMMA_F32_16X16X4_F32.
//
// Per workgroup (1 wave): 64x64 output tile == 16x16 input cells.
//   1) load 20x20 clamped input patch -> LDS
//   2) horizontal: D[outcol(16) x row(16)] = A1(16x4)xB1 + A2(16x4)xB2 -> LDS
//   3) vertical:   D[outrow(16) x col(16)] = A1xB1 + A2xB2 -> global stores
// A-matrix (Keys weights for the 4 phases of the x4 resize) is a per-lane
// constant: for output t in a 16-group, s=(t-1.5)/4, taps at patch cols
// k0..k0+7 with weight keys_w(s + 2 - j).
// ---------------------------------------------------------------------------
__global__ void __launch_bounds__(32)
upsample4_wmma_kernel(const float* __restrict__ h256,
                      float* __restrict__ out, int B)
{
    __shared__ float sIn[32][21];   // 20x20 patch (+pad rows), odd pitch
    __shared__ float sTmp[20][65];  // horizontal result: [in-row][out-col]

    const int lane = threadIdx.x;           // 0..31
    const int tx = blockIdx.x;              // 0..15 tile col
    const int ty = blockIdx.y;              // 0..15 tile row
    const int bz = blockIdx.z;              // batch

    const int ix0 = tx * 16;                // input-tile origin
    const int iy0 = ty * 16;
    const float* src = h256 + (size_t)bz * 256 * 256;
    float* dst = out + (size_t)(B - 1 - bz) * 1024 * 1024;   // batch flip

    // zero patch (incl. pad rows 20..31 so rt=1 tile reads defined data)
    for (int i = lane; i < 32 * 21; i += 32) ((float*)sIn)[i] = 0.0f;
    // load 20x20 edge-clamped patch
    for (int i = lane; i < 400; i += 32) {
        const int py = i / 20, px = i % 20;
        const int gy = min(max(iy0 - 2 + py, 0), 255);
        const int gx = min(max(ix0 - 2 + px, 0), 255);
        sIn[py][px] = src[gy * 256 + gx];
    }

    // ---- A matrix per ISA 32-bit 16x4 layout ----
    // lanes 0-15: M=lane, vgpr0=K0, vgpr1=K1 ; lanes 16-31: K2, K3
    const int m  = lane & 15;
    const int hl = lane >> 4;        // 0 or 1
    const int kb = hl * 2;           // K base within a 4-group
    const float s = ((float)m - 1.5f) * 0.25f;
    v2f a1, a2;
    a1.x = keys_w(s + 2.0f - (float)(kb + 0));
    a1.y = keys_w(s + 2.0f - (float)(kb + 1));
    a2.x = keys_w(s + 2.0f - (float)(kb + 4));
    a2.y = keys_w(s + 2.0f - (float)(kb + 5));

    // ---- horizontal pass: sTmp[row][c*16 + t] ----
    // B[K=patch col][N=image row]: b.x/b.y mirror A's K striping.
    const int row = (0) /*set per rt*/;
    (void)row;
#pragma unroll
    for (int rt = 0; rt < 2; ++rt) {
        const int r = rt * 16 + m;   // N = image row for B-load and D-store
#pragma unroll
        for (int c = 0; c < 4; ++c) {
            v2f b1, b2;
            b1.x = sIn[r][c * 4 + kb + 0];
            b1.y = sIn[r][c * 4 + kb + 1];
            b2.x = sIn[r][c * 4 + kb + 4];
            b2.y = sIn[r][c * 4 + kb + 5];
            v8f acc = {};
            acc = __builtin_amdgcn_wmma_f32_16x16x4_f32(
                false, a1, false, b1, (short)0, acc, false, false);
            acc = __builtin_amdgcn_wmma_f32_16x16x4_f32(
                false, a2, false, b2, (short)0, acc, false, false);
            if (r < 20) {            // rows 20..31 of rt=1 are padding
#pragma unroll
                for (int v = 0; v < 8; ++v)
                    sTmp[r][c * 16 + v + hl * 8] = acc[v];   // M = v + 8*hl
            }
        }
    }

    // ---- vertical pass: D[M=out row][N=out col] -> global ----
    // B[K=tmp row][N=out col]; tmp rows for tile r are r*4 .. r*4+7.
#pragma unroll
    for (int r = 0; r < 4; ++r) {
#pragma unroll
        for (int nt = 0; nt < 4; ++nt) {
            const int col = nt * 16 + m;      // N = out col (contiguous loads)
            v2f b1, b2;
            b1.x = sTmp[r * 4 + kb + 0][col];
            b1.y = sTmp[r * 4 + kb + 1][col];
            b2.x = sTmp[r * 4 + kb + 4][col];
            b2.y = sTmp[r * 4 + kb + 5][col];
            v8f acc = {};
            acc = __builtin_amdgcn_wmma_f32_16x16x4_f32(
                false, a1, false, b1, (short)0, acc, false, false);
            acc = __builtin_amdgcn_wmma_f32_16x16x4_f32(
                false, a2, false, b2, (short)0, acc, false, false);
            const int gx = tx * 64 + nt * 16 + m;
#pragma unroll
            for (int v = 0; v < 8; ++v) {
                const int gy = ty * 64 + r * 16 + v + hl * 8;  // M = v + 8*hl
                dst[(size_t)gy * 1024 + gx] = acc[v];
            }
        }
    }
}

// ---------------------------------------------------------------------------
extern "C" void kernel_launch(void* const* d_in, const int* in_sizes, int n_in,
                              void* d_out, int out_size, void* d_ws, size_t ws_size,
                              hipStream_t stream)
{
    (void)in_sizes; (void)n_in; (void)out_size; (void)ws_size;

    const float* cam256 = (const float*)d_in[0];
    const float* cam128 = (const float*)d_in[1];
    const float* cam64  = (const float*)d_in[2];
    const float* cam32  = (const float*)d_in[3];
    const float* w1 = (const float*)d_in[4];
    const float* b1 = (const float*)d_in[5];
    const float* w2 = (const float*)d_in[6];
    const float* b2 = (const float*)d_in[7];
    const float* w3 = (const float*)d_in[8];
    const float* b3 = (const float*)d_in[9];
    float* out = (float*)d_out;
    const int B = 64;

    float* h64  = (float*)d_ws;                        // B*64*64
    float* h128 = h64  + (size_t)B * 64 * 64;          // B*128*128
    float* h256 = h128 + (size_t)B * 128 * 128;        // B*256*256

    {   // stage 1: cam_32 -> 64, mix cam_64
        const long long t = (long long)B * 64 * 64;
        stage_up2_mix_relu<<<(unsigned)((t + 255) / 256), 256, 0, stream>>>(
            cam32, cam64, w1, b1, h64, 32, B);
    }
    {   // stage 2: h64 -> 128, mix cam_128
        const long long t = (long long)B * 128 * 128;
        stage_up2_mix_relu<<<(unsigned)((t + 255) / 256), 256, 0, stream>>>(
            h64, cam128, w2, b2, h128, 64, B);
    }
    {   // stage 3: h128 -> 256, mix cam_256
        const long long t = (long long)B * 256 * 256;
        stage_up2_mix_relu<<<(unsigned)((t + 255) / 256), 256, 0, stream>>>(
            h128, cam256, w3, b3, h256, 128, B);
    }
    {   // stage 4: h256 -> 1024 via WMMA, batch-flipped
        dim3 grid(16, 16, B);
        upsample4_wmma_kernel<<<grid, 32, 0, stream>>>(h256, out, B);
    }
}